// MHAbias_76647986364552
// MI455X (gfx1250) — compile-verified
//
#include <hip/hip_runtime.h>

typedef __attribute__((ext_vector_type(2))) float v2f;
typedef __attribute__((ext_vector_type(8))) float v8f;

#define LOG2E 1.4426950408889634f

static __device__ __forceinline__ v8f wmma_f32_4(v2f a, v2f b, v8f c) {
  // 8 args: (neg_a, A, neg_b, B, c_mod, C, reuse_a, reuse_b)
  return __builtin_amdgcn_wmma_f32_16x16x4_f32(false, a, false, b, (short)0, c,
                                               false, false);
}

// Single-instruction 16-lane butterfly step: DPP as a source modifier on the
// max itself (v_max_num_f32_dpp dst, src0.dpp, src1) -- no mov, no canonicalize.
#define DPP_MAX_STEP(v, SEL)                                                  \
  asm("v_max_num_f32_dpp %0, %1, %1 " SEL                                     \
      " row_mask:0xf bank_mask:0xf bound_ctrl:1"                              \
      : "=v"(v)                                                               \
      : "v"(v))

// Block = 8 waves = the 8 heads of one (b, 16-row P tile): waves share mask
// addresses and bias cachelines (64B bias line = 2 s x 8 h) through WGP$.
// Each wave: flash-attention over S in 64-column super-tiles with f32 WMMA.
// - scores C layout (rows m in VGPRs, cols s in lanes) keeps the dominant
//   134MB bias stream lane-coalesced.
// - row max: v_max3 combine + 4 one-instruction DPP butterfly steps.
// - row sum: folded into the matrix pipe via ones-B WMMA (lacc, C layout).
// - softmax in base-2: P = exp2(sc*log2e - m*log2e), 2 VALU/element.
// - P tile C->A layout via stride-20 LDS (bank-conflict-free, in-order DS).
// - all gathers are per-lane base + compile-time immediate offsets.
__global__ __launch_bounds__(256, 1) void mha16_kernel(
    const float* __restrict__ q, const float* __restrict__ k,
    const float* __restrict__ v, const float* __restrict__ bias,
    const float* __restrict__ mask, float* __restrict__ out) {
  __shared__ __align__(16) float ldsP[8][320];  // 16 rows x stride 20, per wave

  const int tid  = threadIdx.x;
  const int lane = tid & 31;
  const int wave = tid >> 5;        // = head h
  const int b     = blockIdx.x >> 6;
  const int ptile = blockIdx.x & 63;
  const int h  = wave;
  const int bh = b * 8 + h;
  const int p0 = ptile << 4;

  const int n    = lane & 15;  // N column (C/D, B layouts) and M row (A loads)
  const int half = lane >> 4;

  const float* qp = q + ((size_t)bh * 1024 + p0) * 16;  // [16][16]

  // Per-lane gather bases; inner-loop offsets are all compile-time constants.
  const float* kpl = k + (size_t)bh * 1024 * 16 + n * 16 + 2 * half;
  const float* vpl = v + (size_t)bh * 1024 * 16 + 32 * half + n;
  const float* bpl = bias + (((size_t)b * 1024 + p0) * 1024) * 8 + h +
                     ((size_t)(8 * half) * 1024 + n) * 8;
  const float* mpl = mask + ((size_t)b * 1024 + p0) * 1024 +
                     (size_t)(8 * half) * 1024 + n;

  // Q tile in A layout: lane m = n, VGPR r -> K = 4c + 2*half + r (float2)
  v2f qa[4];
#pragma unroll
  for (int c = 0; c < 4; ++c)
    qa[c] = *(const v2f*)(qp + n * 16 + 4 * c + 2 * half);

  v8f acc  = {0.f, 0.f, 0.f, 0.f, 0.f, 0.f, 0.f, 0.f};
  v8f lacc = {0.f, 0.f, 0.f, 0.f, 0.f, 0.f, 0.f, 0.f};
  float mrow[8], negml2e[8];
#pragma unroll
  for (int r = 0; r < 8; ++r) { mrow[r] = -3.0e38f; negml2e[r] = 3.0e38f; }

  float* myP = ldsP[wave];
  const v2f onesB = {1.0f, 1.0f};

  for (int it = 0; it < 16; ++it) {
    // ---- scores for 4 sub-tiles: B layout lane s = n, VGPR r -> d
    v8f sc[4];
#pragma unroll
    for (int t = 0; t < 4; ++t) {
      v8f z = {0.f, 0.f, 0.f, 0.f, 0.f, 0.f, 0.f, 0.f};
#pragma unroll
      for (int c = 0; c < 4; ++c) {
        v2f kb = *(const v2f*)(kpl + t * 256 + 4 * c);  // (t*16 s-rows)*16 + 4c
        z = wmma_f32_4(qa[c], kb, z);
      }
      sc[t] = z;
    }

    // ---- scale + mask + bias; row m = r + 8*half, col s = s0 + t*16 + n
#pragma unroll
    for (int t = 0; t < 4; ++t) {
#pragma unroll
      for (int r = 0; r < 8; ++r) {
        const float bi = bpl[r * 8192 + t * 128];  // (r*1024 + t*16) * 8
        const float mk = mpl[r * 1024 + t * 16];
        sc[t][r] = fmaf(sc[t][r], 0.25f, mk + bi);  // SCALE = 1/sqrt(16)
      }
    }

    // ---- online softmax: 64-col row max via max3 combine + DPP butterfly
    float pscale[8];
#pragma unroll
    for (int r = 0; r < 8; ++r) {
      float mx = fmaxf(fmaxf(sc[0][r], sc[1][r]), fmaxf(sc[2][r], sc[3][r]));
      DPP_MAX_STEP(mx, "quad_perm:[1,0,3,2]");  // xor 1
      DPP_MAX_STEP(mx, "quad_perm:[2,3,0,1]");  // xor 2
      DPP_MAX_STEP(mx, "row_half_mirror");      // combine 4-groups
      DPP_MAX_STEP(mx, "row_mirror");           // combine 8-groups
      const float mnew = fmaxf(mrow[r], mx);
      const float nml2e = -mnew * LOG2E;
      pscale[r] = __builtin_amdgcn_exp2f(fmaf(mrow[r], LOG2E, nml2e));
      mrow[r] = mnew;
      negml2e[r] = nml2e;
      acc[r]  *= pscale[r];
      lacc[r] *= pscale[r];
    }

    // ---- per sub-tile: exp2 -> LDS transpose -> PV WMMA + ones-WMMA rowsum
#pragma unroll
    for (int t = 0; t < 4; ++t) {
#pragma unroll
      for (int r = 0; r < 8; ++r) {
        const float p =
            __builtin_amdgcn_exp2f(fmaf(sc[t][r], LOG2E, negml2e[r]));
        myP[(r + 8 * half) * 20 + n] = p;
      }
      asm volatile("" ::: "memory");  // cross-lane LDS visibility fence
#pragma unroll
      for (int c = 0; c < 4; ++c) {
        v2f pa = *(const v2f*)(myP + n * 20 + 4 * c + 2 * half);
        v2f vb;
        vb[0] = vpl[t * 256 + 64 * c];       // (t*16 + 4c + 2*half + 0)*16 + n
        vb[1] = vpl[t * 256 + 64 * c + 16];  // (t*16 + 4c + 2*half + 1)*16 + n
        acc  = wmma_f32_4(pa, vb, acc);
        lacc = wmma_f32_4(pa, onesB, lacc);
      }
      asm volatile("" ::: "memory");  // keep next stores below these loads
    }

    // advance per-iteration bases: s0 += 64
    kpl += 64 * 16;
    vpl += 64 * 16;
    bpl += 64 * 8;
    mpl += 64;
  }

  // ---- out[b, p, h*16 + d]: elementwise normalize (lacc = row sums, C layout)
#pragma unroll
  for (int r = 0; r < 8; ++r) {
    const int m = r + 8 * half;
    out[(size_t)(b * 1024 + p0 + m) * 128 + h * 16 + n] = acc[r] / lacc[r];
  }
}

extern "C" void kernel_launch(void* const* d_in, const int* in_sizes, int n_in,
                              void* d_out, int out_size, void* d_ws,
                              size_t ws_size, hipStream_t stream) {
  (void)in_sizes; (void)n_in; (void)out_size; (void)d_ws; (void)ws_size;
  const float* q    = (const float*)d_in[0];
  const float* k    = (const float*)d_in[1];
  const float* v    = (const float*)d_in[2];
  const float* bias = (const float*)d_in[3];
  const float* mask = (const float*)d_in[4];
  float* out = (float*)d_out;
  // 256 blocks = 4 b x 64 p-tiles; 8 waves/block = 8 heads sharing bias lines
  mha16_kernel<<<256, 256, 0, stream>>>(q, k, v, bias, mask, out);
}